// CenterLoss2_40965398069571
// MI455X (gfx1250) — compile-verified
//
#include <hip/hip_runtime.h>
#include <stdint.h>

#define NB_CLASS 100000
#define DIM 256
#define BATCH 16384
#define LOSS_W 0.01f
#define ALPHA_C 0.05f

// ---------------------------------------------------------------------------
// ws layout: [0 .. NB_CLASS) int counts | [NB_CLASS] float loss accumulator
// ---------------------------------------------------------------------------

__global__ void cl_zero_ws(int* __restrict__ counts, float* __restrict__ lossacc) {
  int i = blockIdx.x * blockDim.x + threadIdx.x;
  if (i < NB_CLASS) counts[i] = 0;
  if (i == 0) lossacc[0] = 0.0f;
}

__global__ void cl_count(const int* __restrict__ y, int* __restrict__ counts) {
  int i = blockIdx.x * blockDim.x + threadIdx.x;
  if (i < BATCH) atomicAdd(&counts[y[i]], 1);
}

// Bulk copy centers -> d_out+1. Source is 16B aligned (b128 loads); the
// destination is offset by one float (4B aligned) so store scalar f32 (the
// backend may merge; HW allows dword-aligned b128). Side effect we want:
// pulls the whole 102 MB table into the 192 MB L2 so the later random gather
// + scatter atomics stay in L2.
__global__ void cl_copy_centers(const float* __restrict__ src, float* __restrict__ dst) {
  size_t g = ((size_t)blockIdx.x * blockDim.x + threadIdx.x) * 4;
  float4 v = *(const float4*)(src + g);
  dst[g + 0] = v.x;
  dst[g + 1] = v.y;
  dst[g + 2] = v.z;
  dst[g + 3] = v.w;
}

// One wave32 per sample row; 8 floats per lane as two b128 chunks.
__global__ __launch_bounds__(256) void cl_main(
    const float* __restrict__ x, const int* __restrict__ y,
    const float* __restrict__ centers, const int* __restrict__ counts,
    float* __restrict__ out_centers, float* __restrict__ lossacc) {
  __shared__ float xs[8 * DIM];   // 8 KB: one x row per wave
  __shared__ float partial[8];    // per-wave loss partials
  const int lane = threadIdx.x & 31;
  const int wid  = threadIdx.x >> 5;
  const int row  = blockIdx.x * 8 + wid;   // grid exactly covers BATCH

  const float* gx0 = x + (size_t)row * DIM + lane * 4;
  const float* gx1 = gx0 + 128;
  // addrspace(3) byte offsets for the async destination (low 32 bits of the
  // flat shared address == workgroup-relative LDS offset)
  unsigned lds0 = (unsigned)(uintptr_t)(void*)&xs[wid * DIM + lane * 4];
  unsigned lds1 = (unsigned)(uintptr_t)(void*)&xs[wid * DIM + 128 + lane * 4];

  // CDNA5 async tensor path: global -> LDS with no VGPR round-trip,
  // tracked on ASYNCcnt. Overlaps with the random center-row gather below.
  asm volatile("global_load_async_to_lds_b128 %0, %1, off"
               :: "v"(lds0), "v"(gx0) : "memory");
  asm volatile("global_load_async_to_lds_b128 %0, %1, off"
               :: "v"(lds1), "v"(gx1) : "memory");

  // Class id is wave-uniform: force it into an SGPR so the count becomes a
  // scalar load and the center-row base folds into saddr-form global loads.
  const int c = __builtin_amdgcn_readfirstlane(y[row]);
  const float* crow = centers + (size_t)c * DIM;
  float4 ca = *(const float4*)(crow + lane * 4);
  float4 cb = *(const float4*)(crow + 128 + lane * 4);
  float cnt   = (float)counts[c];
  float scale = ALPHA_C / (cnt + 1e-6f);

  asm volatile("s_wait_asynccnt 0" ::: "memory");
  float4 xa = *(const float4*)&xs[wid * DIM + lane * 4];
  float4 xb = *(const float4*)&xs[wid * DIM + 128 + lane * 4];

  float d0 = ca.x - xa.x, d1 = ca.y - xa.y, d2 = ca.z - xa.z, d3 = ca.w - xa.w;
  float d4 = cb.x - xb.x, d5 = cb.y - xb.y, d6 = cb.z - xb.z, d7 = cb.w - xb.w;

  // Order-independent scatter: non-returning f32 atomics RMW in L2.
  // Label collisions are rare (16K samples over 100K classes) -> no hot lines.
  float* o0 = out_centers + (size_t)c * DIM + lane * 4;
  float* o1 = o0 + 128;
  __hip_atomic_fetch_add(o0 + 0, -scale * d0, __ATOMIC_RELAXED, __HIP_MEMORY_SCOPE_AGENT);
  __hip_atomic_fetch_add(o0 + 1, -scale * d1, __ATOMIC_RELAXED, __HIP_MEMORY_SCOPE_AGENT);
  __hip_atomic_fetch_add(o0 + 2, -scale * d2, __ATOMIC_RELAXED, __HIP_MEMORY_SCOPE_AGENT);
  __hip_atomic_fetch_add(o0 + 3, -scale * d3, __ATOMIC_RELAXED, __HIP_MEMORY_SCOPE_AGENT);
  __hip_atomic_fetch_add(o1 + 0, -scale * d4, __ATOMIC_RELAXED, __HIP_MEMORY_SCOPE_AGENT);
  __hip_atomic_fetch_add(o1 + 1, -scale * d5, __ATOMIC_RELAXED, __HIP_MEMORY_SCOPE_AGENT);
  __hip_atomic_fetch_add(o1 + 2, -scale * d6, __ATOMIC_RELAXED, __HIP_MEMORY_SCOPE_AGENT);
  __hip_atomic_fetch_add(o1 + 3, -scale * d7, __ATOMIC_RELAXED, __HIP_MEMORY_SCOPE_AGENT);

  // Loss partial: wave32 butterfly, then block-level combine so the single
  // accumulator line sees 2048 atomics instead of 16384 (same-address RMWs
  // serialize in the L2 atomic unit -> this was the critical-path hazard).
  float s = d0 * d0 + d1 * d1 + d2 * d2 + d3 * d3 +
            d4 * d4 + d5 * d5 + d6 * d6 + d7 * d7;
  for (int off = 16; off > 0; off >>= 1) s += __shfl_xor(s, off, 32);
  if (lane == 0) partial[wid] = s;
  __syncthreads();
  if (threadIdx.x == 0) {
    float t = partial[0] + partial[1] + partial[2] + partial[3] +
              partial[4] + partial[5] + partial[6] + partial[7];
    __hip_atomic_fetch_add(lossacc, t, __ATOMIC_RELAXED, __HIP_MEMORY_SCOPE_AGENT);
  }
}

__global__ void cl_finalize(const float* __restrict__ lossacc, float* __restrict__ out_loss) {
  // mean over BATCH*DIM = 4194304 (exact power of two)
  out_loss[0] = LOSS_W * lossacc[0] * (1.0f / 4194304.0f);
}

extern "C" void kernel_launch(void* const* d_in, const int* in_sizes, int n_in,
                              void* d_out, int out_size, void* d_ws, size_t ws_size,
                              hipStream_t stream) {
  const float* x       = (const float*)d_in[0];
  const int*   y       = (const int*)d_in[1];
  const float* centers = (const float*)d_in[2];

  float* out_loss    = (float*)d_out;       // output[0]  : scalar loss
  float* out_centers = (float*)d_out + 1;   // output[1..]: new_centers (row-major)

  int*   counts  = (int*)d_ws;
  float* lossacc = (float*)((char*)d_ws + (size_t)NB_CLASS * sizeof(int));

  cl_zero_ws<<<(NB_CLASS + 255) / 256, 256, 0, stream>>>(counts, lossacc);
  cl_count<<<BATCH / 256, 256, 0, stream>>>(y, counts);
  cl_copy_centers<<<(NB_CLASS * DIM / 4) / 256, 256, 0, stream>>>(centers, out_centers);
  cl_main<<<BATCH / 8, 256, 0, stream>>>(x, y, centers, counts, out_centers, lossacc);
  cl_finalize<<<1, 1, 0, stream>>>(lossacc, out_loss);
}